// LSTMWithWordAttention_27350351741654
// MI455X (gfx1250) — compile-verified
//
#include <hip/hip_runtime.h>
#include <math.h>

// CDNA5 (gfx1250) wave32 WMMA, fp32 16x16x4.
typedef __attribute__((ext_vector_type(2))) float v2f;
typedef __attribute__((ext_vector_type(8))) float v8f;

__device__ __forceinline__ v8f wmma4(v2f a, v2f b, v8f c) {
  // D = A(16x4,f32) * B(4x16,f32) + C(16x16,f32)
  return __builtin_amdgcn_wmma_f32_16x16x4_f32(false, a, false, b, (short)0, c,
                                               false, false);
}

__device__ __forceinline__ float sigmoidf_(float x) {
  return 1.0f / (1.0f + __expf(-x));
}

// ---------------------------------------------------------------------------
// Generic C[M,N] = A[M,K] @ W[N,K]^T (+ b1 + b2).  A rows either direct or
// gathered from the embedding table (tokens != nullptr).  Each wave computes a
// 16x32 strip (two 16x16 tiles sharing one A load); a 256-thread block covers
// 16(M) x 256(N).
// ---------------------------------------------------------------------------
__global__ void __launch_bounds__(256)
wmma_gemm_f32(const float* __restrict__ A, const int* __restrict__ tokens,
              const float* __restrict__ emb, const float* __restrict__ W,
              const float* __restrict__ b1, const float* __restrict__ b2,
              float* __restrict__ C, int K, int N) {
  const int lane = threadIdx.x & 31;
  const int wav  = threadIdx.x >> 5;
  const int m0   = blockIdx.y << 4;
  const int n0   = (blockIdx.x << 8) + (wav << 5);
  const int r16  = lane & 15;
  const int kh   = (lane >> 4) << 1;   // K-half select: lanes 16..31 -> K+2

  const float* arow;
  if (tokens)
    arow = emb + (size_t)tokens[m0 + r16] * K;   // fused embedding gather
  else
    arow = A + (size_t)(m0 + r16) * K;
  const float* wrow0 = W + (size_t)(n0 + r16) * K;
  const float* wrow1 = W + (size_t)(n0 + 16 + r16) * K;

  v8f acc0 = {}, acc1 = {};
  for (int k = 0; k < K; k += 4) {
    v2f a  = *(const v2f*)(arow + k + kh);      // shared across both N-tiles
    v2f b0 = *(const v2f*)(wrow0 + k + kh);
    v2f b1 = *(const v2f*)(wrow1 + k + kh);
    acc0 = wmma4(a, b0, acc0);
    acc1 = wmma4(a, b1, acc1);
  }

  const int nA = n0 + r16;
  const int nB = n0 + 16 + r16;
  float biasA = 0.f, biasB = 0.f;
  if (b1) { biasA += b1[nA]; biasB += b1[nB]; }
  if (b2) { biasA += b2[nA]; biasB += b2[nB]; }
  const int mo = m0 + ((lane >> 4) << 3);
  float* crow = C + (size_t)mo * N;
#pragma unroll
  for (int j = 0; j < 8; ++j) {
    crow[(size_t)j * N + nA] = acc0[j] + biasA;
    crow[(size_t)j * N + nB] = acc1[j] + biasB;
  }
}

// ---------------------------------------------------------------------------
// One LSTM timestep: g = Gx[t] + h_prev @ Whh^T, then gates.  Grid = 32 blocks
// (one per 16 h-columns).  h_prev (64x512 = 128 KB) is staged once per block
// into LDS via the gfx1250 async-to-LDS engine, then all 8 waves feed their
// WMMA A-operands from LDS.  Each wave owns one batch tile and two gates
// (g, g+2) sharing every A fetch; results staged in LDS, nonlinearity applied
// locally.  h/c are ping-ponged across steps by the host.
// ---------------------------------------------------------------------------
__global__ void __launch_bounds__(256)
lstm_step_kernel(const float* __restrict__ hprev, const float* __restrict__ cprev,
                 const float* __restrict__ Whh, const float* __restrict__ Gx,
                 float* __restrict__ hout, float* __restrict__ cout,
                 float* __restrict__ seq_out, int t, int T) {
  __shared__ float hsh[64 * 512];      // staged h_prev (128 KB)
  __shared__ float glds[4][64][16];    // [gate][batch][h-local] (16 KB)
  const int tid  = threadIdx.x;
  const int lane = tid & 31;
  const int wav  = tid >> 5;
  const int n0   = blockIdx.x << 4;
  const int r16  = lane & 15;
  const int kh   = (lane >> 4) << 1;

  // --- async stage of h_prev into LDS (tracked by ASYNCcnt) ---
  {
    const unsigned lds0 = (unsigned)(size_t)(&hsh[0]);  // LDS byte offset
    for (int i = tid; i < (64 * 512) / 4; i += 256) {
      unsigned long long ga = (unsigned long long)(const void*)(hprev + i * 4);
      unsigned la = lds0 + (unsigned)(i * 16);
      asm volatile("global_load_async_to_lds_b128 %0, %1, off"
                   :: "v"(la), "v"(ga) : "memory");
    }
    asm volatile("s_wait_asynccnt 0x0" ::: "memory");
  }
  __syncthreads();

  const int mt = wav & 3;              // batch tile (4 x 16 rows)
  const int g0 = wav >> 2;             // gates g0 and g0+2
  const float* arow = &hsh[((mt << 4) + r16) * 512];
  const float* w0 = Whh + (size_t)((g0 << 9) + n0 + r16) * 512;
  const float* w1 = Whh + (size_t)(((g0 + 2) << 9) + n0 + r16) * 512;
  v8f acc0 = {}, acc1 = {};
  for (int k = 0; k < 512; k += 4) {
    v2f a  = *(const v2f*)(arow + k + kh);     // ds_load_b64 (LDS)
    v2f b0 = *(const v2f*)(w0 + k + kh);
    v2f b1 = *(const v2f*)(w1 + k + kh);
    acc0 = wmma4(a, b0, acc0);
    acc1 = wmma4(a, b1, acc1);
  }
  const int mbase = (mt << 4) + ((lane >> 4) << 3);
#pragma unroll
  for (int j = 0; j < 8; ++j) {
    glds[g0][mbase + j][r16]     = acc0[j];
    glds[g0 + 2][mbase + j][r16] = acc1[j];
  }
  __syncthreads();

  for (int idx = tid; idx < 1024; idx += 256) {
    const int m    = idx >> 4;
    const int nl   = idx & 15;
    const int hcol = n0 + nl;
    const size_t gxb = ((size_t)m * T + t) * 2048 + hcol;
    float gi = glds[0][m][nl] + Gx[gxb];
    float gf = glds[1][m][nl] + Gx[gxb + 512];
    float gg = glds[2][m][nl] + Gx[gxb + 1024];
    float go = glds[3][m][nl] + Gx[gxb + 1536];
    float c = sigmoidf_(gf) * cprev[m * 512 + hcol] + sigmoidf_(gi) * tanhf(gg);
    float h = sigmoidf_(go) * tanhf(c);
    cout[m * 512 + hcol] = c;
    hout[m * 512 + hcol] = h;
    seq_out[((size_t)m * T + t) * 512 + hcol] = h;
  }
}

// ---------------------------------------------------------------------------
// Attention per-step projections: z=0: a2 = q_t@W_h^T + r@W_r^T
//                                 z=1: rt = tanh(r@W_t^T)
// One wave per 16x16 tile of the [64,512] outputs.
// ---------------------------------------------------------------------------
__global__ void __launch_bounds__(32)
att_pre_kernel(const float* __restrict__ outh, int t, const float* __restrict__ r,
               const float* __restrict__ Wh, const float* __restrict__ Wr,
               const float* __restrict__ Wt, float* __restrict__ a2,
               float* __restrict__ rt) {
  const int lane = threadIdx.x & 31;
  const int m0   = blockIdx.x << 4;
  const int n0   = blockIdx.y << 4;
  const int r16  = lane & 15;
  const int kh   = (lane >> 4) << 1;
  const int mo   = m0 + ((lane >> 4) << 3);
  v8f acc = {};
  if (blockIdx.z == 0) {
    const float* aq = outh + ((size_t)(m0 + r16) * 64 + t) * 512;   // q_t rows
    const float* w1 = Wh + (size_t)(n0 + r16) * 512;
    for (int k = 0; k < 512; k += 4)
      acc = wmma4(*(const v2f*)(aq + k + kh), *(const v2f*)(w1 + k + kh), acc);
    const float* ar = r + (size_t)(m0 + r16) * 512;
    const float* w2 = Wr + (size_t)(n0 + r16) * 512;
    for (int k = 0; k < 512; k += 4)
      acc = wmma4(*(const v2f*)(ar + k + kh), *(const v2f*)(w2 + k + kh), acc);
#pragma unroll
    for (int j = 0; j < 8; ++j)
      a2[(size_t)(mo + j) * 512 + n0 + r16] = acc[j];
  } else {
    const float* ar = r + (size_t)(m0 + r16) * 512;
    const float* w3 = Wt + (size_t)(n0 + r16) * 512;
    for (int k = 0; k < 512; k += 4)
      acc = wmma4(*(const v2f*)(ar + k + kh), *(const v2f*)(w3 + k + kh), acc);
#pragma unroll
    for (int j = 0; j < 8; ++j)
      rt[(size_t)(mo + j) * 512 + n0 + r16] = tanhf(acc[j]);
  }
}

// ---------------------------------------------------------------------------
// Attention scores + softmax + context per batch row (one block per b):
//  s[p]   = sum_h w[h]*tanh(a1t[b,p,h] + a2[b,h])       (wave-parallel over h)
//  sc     = softmax_p(s)
//  r[b,h] = sum_p outp[b,p,h]*sc[p] + rt[b,h]
// ---------------------------------------------------------------------------
__global__ void __launch_bounds__(256)
att_score_kernel(const float* __restrict__ a1t, const float* __restrict__ a2,
                 const float* __restrict__ watt, const float* __restrict__ outp,
                 const float* __restrict__ rt, float* __restrict__ r) {
  __shared__ float s_lds[128];
  __shared__ float red[128];
  const int b    = blockIdx.x;
  const int tid  = threadIdx.x;
  const int lane = tid & 31;
  const int wav  = tid >> 5;
  const float* a2b = a2 + (size_t)b * 512;

  for (int p = wav; p < 128; p += 8) {
    const float* a1row = a1t + ((size_t)b * 128 + p) * 512;
    float part = 0.f;
    for (int h = lane; h < 512; h += 32)
      part += watt[h] * tanhf(a1row[h] + a2b[h]);
#pragma unroll
    for (int off = 16; off >= 1; off >>= 1) part += __shfl_xor(part, off, 32);
    if (lane == 0) s_lds[p] = part;
  }
  __syncthreads();

  if (tid < 128) red[tid] = s_lds[tid];
  __syncthreads();
  for (int off = 64; off >= 1; off >>= 1) {
    if (tid < off) red[tid] = fmaxf(red[tid], red[tid + off]);
    __syncthreads();
  }
  const float mx = red[0];
  __syncthreads();
  float e = 0.f;
  if (tid < 128) { e = __expf(s_lds[tid] - mx); red[tid] = e; }
  __syncthreads();
  for (int off = 64; off >= 1; off >>= 1) {
    if (tid < off) red[tid] += red[tid + off];
    __syncthreads();
  }
  const float inv = 1.0f / red[0];
  __syncthreads();
  if (tid < 128) s_lds[tid] = e * inv;
  __syncthreads();

  for (int h = tid; h < 512; h += 256) {
    const float* op = outp + (size_t)b * 128 * 512 + h;
    float accv = 0.f;
    for (int p = 0; p < 128; ++p) accv += op[(size_t)p * 512] * s_lds[p];
    r[(size_t)b * 512 + h] = accv + rt[(size_t)b * 512 + h];
  }
}

// rep = tanh(r@fc1^T + fc1_b + hn@fc2^T + fc2_b), one wave per 16x16 tile.
__global__ void __launch_bounds__(32)
final_rep_kernel(const float* __restrict__ r, const float* __restrict__ hn,
                 const float* __restrict__ fc1w, const float* __restrict__ fc1b,
                 const float* __restrict__ fc2w, const float* __restrict__ fc2b,
                 float* __restrict__ rep) {
  const int lane = threadIdx.x & 31;
  const int m0   = blockIdx.x << 4;
  const int n0   = blockIdx.y << 4;
  const int r16  = lane & 15;
  const int kh   = (lane >> 4) << 1;
  v8f acc = {};
  const float* a1 = r + (size_t)(m0 + r16) * 512;
  const float* w1 = fc1w + (size_t)(n0 + r16) * 512;
  for (int k = 0; k < 512; k += 4)
    acc = wmma4(*(const v2f*)(a1 + k + kh), *(const v2f*)(w1 + k + kh), acc);
  const float* a2p = hn + (size_t)(m0 + r16) * 512;
  const float* w2 = fc2w + (size_t)(n0 + r16) * 512;
  for (int k = 0; k < 512; k += 4)
    acc = wmma4(*(const v2f*)(a2p + k + kh), *(const v2f*)(w2 + k + kh), acc);
  const int n = n0 + r16;
  const float bias = fc1b[n] + fc2b[n];
  const int mo = m0 + ((lane >> 4) << 3);
#pragma unroll
  for (int j = 0; j < 8; ++j)
    rep[(size_t)(mo + j) * 512 + n] = tanhf(acc[j] + bias);
}

// out[64,3] = rep @ fc3^T + fc3_b   (too skinny for WMMA; 192 dot products)
__global__ void __launch_bounds__(256)
final_out_kernel(const float* __restrict__ rep, const float* __restrict__ fc3w,
                 const float* __restrict__ fc3b, float* __restrict__ out) {
  const int idx = threadIdx.x;
  if (idx >= 192) return;
  const int b = idx / 3, c = idx % 3;
  const float* rr = rep + (size_t)b * 512;
  const float* wc = fc3w + (size_t)c * 512;
  float s = fc3b[c];
  for (int k = 0; k < 512; ++k) s += rr[k] * wc[k];
  out[idx] = s;
}

__global__ void fill0_kernel(float* __restrict__ p, int n) {
  int i = blockIdx.x * 256 + threadIdx.x;
  if (i < n) p[i] = 0.f;
}

// ---------------------------------------------------------------------------
extern "C" void kernel_launch(void* const* d_in, const int* in_sizes, int n_in,
                              void* d_out, int out_size, void* d_ws,
                              size_t ws_size, hipStream_t stream) {
  (void)in_sizes; (void)n_in; (void)out_size; (void)ws_size;
  const int*   premise = (const int*)d_in[0];
  const int*   hypo    = (const int*)d_in[1];
  const float* emb  = (const float*)d_in[2];
  const float* Wih1 = (const float*)d_in[3];
  const float* Whh1 = (const float*)d_in[4];
  const float* bih1 = (const float*)d_in[5];
  const float* bhh1 = (const float*)d_in[6];
  const float* Wih2 = (const float*)d_in[7];
  const float* Whh2 = (const float*)d_in[8];
  const float* bih2 = (const float*)d_in[9];
  const float* bhh2 = (const float*)d_in[10];
  const float* W_y  = (const float*)d_in[11];
  const float* W_h  = (const float*)d_in[12];
  const float* W_r  = (const float*)d_in[13];
  const float* W_t  = (const float*)d_in[14];
  const float* watt = (const float*)d_in[15];
  const float* fc1w = (const float*)d_in[16];
  const float* fc1b = (const float*)d_in[17];
  const float* fc2w = (const float*)d_in[18];
  const float* fc2b = (const float*)d_in[19];
  const float* fc3w = (const float*)d_in[20];
  const float* fc3b = (const float*)d_in[21];

  float* ws = (float*)d_ws;
  size_t off = 0;
  auto take = [&](size_t n) { float* p = ws + off; off += n; return p; };
  float* Gxp  = take((size_t)8192 * 2048);   // premise x@Wih^T + bias (hoisted)
  float* Gxh  = take((size_t)4096 * 2048);
  float* outp = take((size_t)8192 * 512);    // premise hidden states [B,P,H]
  float* outh = take((size_t)4096 * 512);    // hypothesis hidden states [B,T,H]
  float* a1t  = take((size_t)8192 * 512);    // a1 transposed: [B,P,H]
  float* hb0 = take(64 * 512); float* hb1 = take(64 * 512);
  float* cb0 = take(64 * 512); float* cb1 = take(64 * 512);
  float* a2   = take(64 * 512);
  float* rt   = take(64 * 512);
  float* rbuf = take(64 * 512);
  float* rep  = take(64 * 512);

  // Hoisted input projections (embedding gather fused into A-load).
  wmma_gemm_f32<<<dim3(8, 512), 256, 0, stream>>>(nullptr, premise, emb, Wih1,
                                                  bih1, bhh1, Gxp, 300, 2048);
  wmma_gemm_f32<<<dim3(8, 256), 256, 0, stream>>>(nullptr, hypo, emb, Wih2,
                                                  bih2, bhh2, Gxh, 300, 2048);

  const int HB = 64 * 512;
  fill0_kernel<<<(HB + 255) / 256, 256, 0, stream>>>(hb0, HB);
  fill0_kernel<<<(HB + 255) / 256, 256, 0, stream>>>(cb0, HB);

  float* hping[2] = {hb0, hb1};
  float* cping[2] = {cb0, cb1};

  // Premise LSTM: 128 sequential steps (ping-pong h/c).
  for (int t = 0; t < 128; ++t)
    lstm_step_kernel<<<32, 256, 0, stream>>>(hping[t & 1], cping[t & 1], Whh1,
                                             Gxp, hping[(t + 1) & 1],
                                             cping[(t + 1) & 1], outp, t, 128);
  // After 128 steps final c_p sits in cb0 (parity 0).

  // a1 (transposed layout [B,P,H]) = outp_flat @ W_y^T — one batched GEMM.
  wmma_gemm_f32<<<dim3(2, 512), 256, 0, stream>>>(outp, nullptr, nullptr, W_y,
                                                  nullptr, nullptr, a1t, 512, 512);

  // Hypothesis LSTM: h0 = 0, c0 = c_p (already in cb0 — just zero h).
  fill0_kernel<<<(HB + 255) / 256, 256, 0, stream>>>(hb0, HB);
  for (int t = 0; t < 64; ++t)
    lstm_step_kernel<<<32, 256, 0, stream>>>(hping[t & 1], cping[t & 1], Whh2,
                                             Gxh, hping[(t + 1) & 1],
                                             cping[(t + 1) & 1], outh, t, 64);
  // hn_h in hb0 (parity 0 after 64 steps).

  // Word-by-word attention: 64 sequential steps.
  fill0_kernel<<<(HB + 255) / 256, 256, 0, stream>>>(rbuf, HB);
  for (int t = 0; t < 64; ++t) {
    att_pre_kernel<<<dim3(4, 32, 2), 32, 0, stream>>>(outh, t, rbuf, W_h, W_r,
                                                      W_t, a2, rt);
    att_score_kernel<<<64, 256, 0, stream>>>(a1t, a2, watt, outp, rt, rbuf);
  }

  final_rep_kernel<<<dim3(4, 32), 32, 0, stream>>>(rbuf, hb0, fc1w, fc1b, fc2w,
                                                   fc2b, rep);
  final_out_kernel<<<1, 256, 0, stream>>>(rep, fc3w, fc3b, (float*)d_out);
}